// ChartParametrizationAD_52020643889814
// MI455X (gfx1250) — compile-verified
//
#include <hip/hip_runtime.h>

// ---------------------------------------------------------------------------
// ChartParametrizationAD: Lam = 50 Stein iterations in sum form.
//   G = I + V^H V ;  G = R^H R (Cholesky-QR)  =>  A = R^-1, C = V R^-1
//   RHS = A^H (V^H Y);  Lam_1 = RHS; Lam <- A Lam W + RHS  (x49)
// Dense math: V_WMMA_F32_16X16X4_F32 (wave32 FP32 WMMA).
// Tile staging: Tensor Data Mover (tensor_load_to_lds, TENSORcnt),
// double-buffered, with hardware LDS padding for bank-conflict avoidance.
// ---------------------------------------------------------------------------

#define NN 512
#define PP 128

typedef __attribute__((ext_vector_type(2))) float v2f;
typedef __attribute__((ext_vector_type(8))) float v8f;
typedef __attribute__((ext_vector_type(4))) unsigned int u32x4;
typedef __attribute__((ext_vector_type(8))) unsigned int u32x8;

__device__ __forceinline__ v8f wmma4(v2f a, v2f b, v8f c) {
  // D = A(16x4,f32) x B(4x16,f32) + C(16x16,f32)
  return __builtin_amdgcn_wmma_f32_16x16x4_f32(
      /*neg_a=*/false, a, /*neg_b=*/false, b,
      /*c_mod=*/(short)0, c, /*reuse_a=*/false, /*reuse_b=*/false);
}

// ---- Tensor Data Mover: 2-D tile load, global -> LDS, with LDS padding ----
// D# per CDNA5 ISA. tensor dims == tile dims (no OOB).
__device__ __forceinline__ void tdm_load_2d(unsigned long long gaddr,
                                            unsigned int lds_byte,
                                            unsigned int tile_x,
                                            unsigned int tile_y,
                                            unsigned int stride_elems,
                                            unsigned int pad_interval,
                                            unsigned int pad_amount) {
  u32x4 g0;
  g0[0] = 1u;                                        // count=1, user mode
  g0[1] = lds_byte;                                  // lds_addr (bytes)
  g0[2] = (unsigned int)(gaddr & 0xFFFFFFFFu);       // global_addr[31:0]
  g0[3] = (unsigned int)((gaddr >> 32) & 0x1FFFFFFu) // global_addr[56:32]
          | (2u << 30);                              // type = 2 ("image")
  u32x8 g1;
  g1[0] = (2u << 16)                                 // data_size = 4 bytes
          | (1u << 20)                               // pad_enable
          | (pad_interval << 22)                     // 2^(i+1) DWORDs per row
          | (pad_amount << 25);                      // (n+1) pad DWORDs
  g1[1] = (tile_x & 0xFFFFu) << 16;                  // tensor_dim0[15:0]
  g1[2] = ((tile_x >> 16) & 0xFFFFu)                 // tensor_dim0[31:16]
          | ((tile_y & 0xFFFFu) << 16);              // tensor_dim1[15:0]
  g1[3] = ((tile_y >> 16) & 0xFFFFu)                 // tensor_dim1[31:16]
          | ((tile_x & 0xFFFFu) << 16);              // tile_dim0
  g1[4] = (tile_y & 0xFFFFu);                        // tile_dim1 (tile_dim2=0)
  g1[5] = stride_elems;                              // tensor_dim0_stride[31:0]
  g1[6] = 0u;                                        // stride hi / dim1_stride lo
  g1[7] = 0u;
  asm volatile("tensor_load_to_lds %0, %1" ::"s"(g0), "s"(g1) : "memory");
}

// LDS plane geometry (floats)
#define APL 1088  // max(64*17 [A normal], 16*66 [A transposed])
#define BPL 1056  // 16*66
#define BUFSZ (2 * APL + 2 * BPL)  // one double-buffer slot: 4288 floats

// Issue the 4 tile-plane DMAs for K-slice k0 into buffer `buf`.
// Waves 0..3 each own one plane, so each wave tracks 1 TENSORcnt op.
template <bool TRA>
__device__ __forceinline__ void issue_tiles(int wid, unsigned int lds_base,
                                            int buf, const float* Ar,
                                            const float* Ai, int lda, int m0,
                                            const float* Br, const float* Bi,
                                            int ldb, int n0, int k0) {
  if (wid >= 4) return;
  unsigned long long ga;
  unsigned int off, tx, ty, stride, pint, pamt;
  if (wid < 2) {
    const float* src = wid ? Ai : Ar;
    off = lds_base + (unsigned int)(buf * BUFSZ + wid * APL) * 4u;
    if (!TRA) {  // A is (M x K): tile 64 rows x 16 cols -> LDS [64][17]
      ga = (unsigned long long)(uintptr_t)(src + (size_t)m0 * lda + k0);
      tx = 16; ty = 64; pint = 3; pamt = 0;  // 16-DWORD rows + 1 pad DWORD
    } else {     // A is (K x M): tile 16 rows x 64 cols -> LDS [16][66]
      ga = (unsigned long long)(uintptr_t)(src + (size_t)k0 * lda + m0);
      tx = 64; ty = 16; pint = 5; pamt = 1;  // 64-DWORD rows + 2 pad DWORDs
    }
    stride = (unsigned int)lda;
  } else {       // B is (K x N): tile 16 rows x 64 cols -> LDS [16][66]
    const float* src = (wid == 3) ? Bi : Br;
    off = lds_base + (unsigned int)(buf * BUFSZ + 2 * APL + (wid - 2) * BPL) * 4u;
    ga = (unsigned long long)(uintptr_t)(src + (size_t)k0 * ldb + n0);
    tx = 64; ty = 16; stride = (unsigned int)ldb; pint = 5; pamt = 1;
  }
  tdm_load_2d(ga, off, tx, ty, stride, pint, pamt);
}

// Complex GEMM: D = opA(A) * B (+ C), matrices as separate re/im planes.
// TRA: opA(A) = A^H with A stored (K x M) row-major; else A is (M x K).
// Conjugation is folded into the epilogue signs (TDM copies raw planes).
// Grid: (N/64, M/64); 256 threads = 8 wave32 waves; K multiple of 16.
template <bool TRA, bool HASC>
__global__ __launch_bounds__(256) void zgemm_wmma(
    const float* __restrict__ Ar, const float* __restrict__ Ai, int lda,
    const float* __restrict__ Br, const float* __restrict__ Bi, int ldb,
    const float* __restrict__ Cr, const float* __restrict__ Ci,
    float* __restrict__ Dr, float* __restrict__ Di, int ldd, int K) {
  __shared__ float smem[2 * BUFSZ];

  const int tid = threadIdx.x;
  const int wid = tid >> 5;
  const int lane = tid & 31;
  const int half = lane >> 4;  // wave32 half: lanes 0-15 / 16-31
  const int lm = lane & 15;
  const int tm = wid >> 1;            // 16-row band of the 64x64 tile
  const int tn0 = (wid & 1) << 1;     // this wave owns n-tiles tn0, tn0+1

  const int m0 = blockIdx.y * 64;
  const int n0 = blockIdx.x * 64;
  const unsigned int lds_base = (unsigned int)(uintptr_t)smem;

  v8f rr0{}, ii0{}, ri0{}, ir0{};
  v8f rr1{}, ii1{}, ri1{}, ir1{};

  const int nk = K >> 4;
  // Prologue: DMA first K-slice into buffer 0.
  issue_tiles<TRA>(wid, lds_base, 0, Ar, Ai, lda, m0, Br, Bi, ldb, n0, 0);

  for (int kt = 0; kt < nk; ++kt) {
    const int buf = kt & 1;
    // Prefetch next K-slice into the other buffer via TDM (overlaps compute).
    if (kt + 1 < nk)
      issue_tiles<TRA>(wid, lds_base, buf ^ 1, Ar, Ai, lda, m0, Br, Bi, ldb,
                       n0, (kt + 1) << 4);
    if (wid < 4) {  // issuing waves: in-order TENSORcnt -> <=1 means cur done
      if (kt + 1 < nk) __builtin_amdgcn_s_wait_tensorcnt(1);
      else             __builtin_amdgcn_s_wait_tensorcnt(0);
    }
    __syncthreads();

    const float* As_r = smem + buf * BUFSZ;
    const float* As_i = As_r + APL;
    const float* Bs_r = smem + buf * BUFSZ + 2 * APL;
    const float* Bs_i = Bs_r + BPL;

#pragma unroll
    for (int q = 0; q < 4; ++q) {
      const int ka = (q << 2) + (half << 1);  // VGPR0 K-index, this half-wave
      const int am = tm * 16 + lm;
      const int cn0 = tn0 * 16 + lm;
      const int cn1 = cn0 + 16;
      v2f ar, ai2, br0, bi0, br1, bi1;
      if (!TRA) {  // A tile [64][17]
        ar.x = As_r[am * 17 + ka];  ar.y = As_r[am * 17 + ka + 1];
        ai2.x = As_i[am * 17 + ka]; ai2.y = As_i[am * 17 + ka + 1];
      } else {     // A tile [16][66] (K-major, like B)
        ar.x = As_r[ka * 66 + am];  ar.y = As_r[(ka + 1) * 66 + am];
        ai2.x = As_i[ka * 66 + am]; ai2.y = As_i[(ka + 1) * 66 + am];
      }
      br0.x = Bs_r[ka * 66 + cn0]; br0.y = Bs_r[(ka + 1) * 66 + cn0];
      bi0.x = Bs_i[ka * 66 + cn0]; bi0.y = Bs_i[(ka + 1) * 66 + cn0];
      br1.x = Bs_r[ka * 66 + cn1]; br1.y = Bs_r[(ka + 1) * 66 + cn1];
      bi1.x = Bs_i[ka * 66 + cn1]; bi1.y = Bs_i[(ka + 1) * 66 + cn1];
      rr0 = wmma4(ar, br0, rr0);
      ii0 = wmma4(ai2, bi0, ii0);
      ri0 = wmma4(ar, bi0, ri0);
      ir0 = wmma4(ai2, br0, ir0);
      rr1 = wmma4(ar, br1, rr1);
      ii1 = wmma4(ai2, bi1, ii1);
      ri1 = wmma4(ar, bi1, ri1);
      ir1 = wmma4(ai2, br1, ir1);
    }
    __syncthreads();  // everyone done with `buf` before it is DMA-overwritten
  }

  // Epilogue. Normal: Re = rr - ii, Im = ri + ir.
  // Conjugate-A:      Re = rr + ii, Im = ri - ir.
#pragma unroll
  for (int r = 0; r < 8; ++r) {
    const int row = m0 + tm * 16 + r + (half << 3);
    const int col0 = n0 + tn0 * 16 + lm;
    const int col1 = col0 + 16;
    float dr0 = TRA ? rr0[r] + ii0[r] : rr0[r] - ii0[r];
    float di0 = TRA ? ri0[r] - ir0[r] : ri0[r] + ir0[r];
    float dr1 = TRA ? rr1[r] + ii1[r] : rr1[r] - ii1[r];
    float di1 = TRA ? ri1[r] - ir1[r] : ri1[r] + ir1[r];
    if (HASC) {
      dr0 += Cr[(size_t)row * ldd + col0]; di0 += Ci[(size_t)row * ldd + col0];
      dr1 += Cr[(size_t)row * ldd + col1]; di1 += Ci[(size_t)row * ldd + col1];
    }
    Dr[(size_t)row * ldd + col0] = dr0; Di[(size_t)row * ldd + col0] = di0;
    Dr[(size_t)row * ldd + col1] = dr1; Di[(size_t)row * ldd + col1] = di1;
  }
}

// G(diag) += 1
__global__ void add_eye_k(float* Gr, int n) {
  int i = blockIdx.x * blockDim.x + threadIdx.x;
  if (i < n) Gr[(size_t)i * n + i] += 1.0f;
}

// In-place complex Cholesky of Hermitian G (upper): G = R^H R.
__global__ __launch_bounds__(256) void chol_k(float* Gr, float* Gi, int n) {
  __shared__ float d;
  for (int k = 0; k < n; ++k) {
    if (threadIdx.x == 0) {
      float v = Gr[(size_t)k * n + k];
      d = sqrtf(fmaxf(v, 1e-30f));
      Gr[(size_t)k * n + k] = d;
      Gi[(size_t)k * n + k] = 0.0f;
    }
    __syncthreads();
    const float dv = d;
    for (int j = k + 1 + threadIdx.x; j < n; j += blockDim.x) {
      Gr[(size_t)k * n + j] /= dv;
      Gi[(size_t)k * n + j] /= dv;
    }
    __syncthreads();
    for (int i = k + 1; i < n; ++i) {
      const float cr = Gr[(size_t)k * n + i];
      const float ci = -Gi[(size_t)k * n + i];  // conj(R[k][i])
      for (int j = i + threadIdx.x; j < n; j += blockDim.x) {
        const float br = Gr[(size_t)k * n + j];
        const float bi = Gi[(size_t)k * n + j];
        Gr[(size_t)i * n + j] -= cr * br - ci * bi;
        Gi[(size_t)i * n + j] -= cr * bi + ci * br;
      }
    }
    __syncthreads();
  }
}

// A = R^-1, upper-triangular complex R, real positive diagonal.
// One thread per column; A accesses coalesce across columns.
__global__ __launch_bounds__(256) void trinv_k(const float* __restrict__ Rr,
                                               const float* __restrict__ Ri,
                                               float* __restrict__ Ar,
                                               float* __restrict__ Ai, int n) {
  const int j = blockIdx.x * blockDim.x + threadIdx.x;
  if (j >= n) return;
  for (int i = j + 1; i < n; ++i) {
    Ar[(size_t)i * n + j] = 0.0f;
    Ai[(size_t)i * n + j] = 0.0f;
  }
  Ar[(size_t)j * n + j] = 1.0f / Rr[(size_t)j * n + j];
  Ai[(size_t)j * n + j] = 0.0f;
  for (int i = j - 1; i >= 0; --i) {
    float sr = 0.0f, si = 0.0f;
    for (int k = i + 1; k <= j; ++k) {
      const float rr = Rr[(size_t)i * n + k];
      const float ri = Ri[(size_t)i * n + k];
      const float xr = Ar[(size_t)k * n + j];
      const float xi = Ai[(size_t)k * n + j];
      sr += rr * xr - ri * xi;
      si += rr * xi + ri * xr;
    }
    const float inv = 1.0f / Rr[(size_t)i * n + i];
    Ar[(size_t)i * n + j] = -sr * inv;
    Ai[(size_t)i * n + j] = -si * inv;
  }
}

__global__ void copy2_k(const float* __restrict__ sr, const float* __restrict__ si,
                        float* __restrict__ dr, float* __restrict__ di, int n) {
  const int i = blockIdx.x * blockDim.x + threadIdx.x;
  if (i < n) { dr[i] = sr[i]; di[i] = si[i]; }
}

__global__ void interleave_k(const float* __restrict__ re,
                             const float* __restrict__ im,
                             float* __restrict__ out, int n) {
  const int i = blockIdx.x * blockDim.x + threadIdx.x;
  if (i < n) {
    out[2 * (size_t)i + 0] = re[i];
    out[2 * (size_t)i + 1] = im[i];
  }
}

extern "C" void kernel_launch(void* const* d_in, const int* in_sizes, int n_in,
                              void* d_out, int out_size, void* d_ws, size_t ws_size,
                              hipStream_t stream) {
  (void)in_sizes; (void)n_in; (void)out_size; (void)ws_size;
  const float* Vp = (const float*)d_in[0];
  const float* Wr = (const float*)d_in[1];
  const float* Wi = (const float*)d_in[2];
  const float* Yr = (const float*)d_in[3];
  const float* Yi = (const float*)d_in[4];
  const float* Vr = Vp;                    // V = P x N, real plane
  const float* Vi = Vp + (size_t)NN * PP;  // imag plane

  float* ws = (float*)d_ws;
  const size_t S = (size_t)NN * NN;
  float* Rr = ws + 0 * S; float* Ri = ws + 1 * S;  // Gram -> Cholesky R
  float* Ar = ws + 2 * S; float* Ai = ws + 3 * S;  // A = R^-1
  float* Tr = ws + 4 * S; float* Ti = ws + 5 * S;  // U, then T = A*Lam
  float* Hr = ws + 6 * S; float* Hi = ws + 7 * S;  // RHS
  float* Lr = ws + 8 * S; float* Li = ws + 9 * S;  // Lam

  const dim3 blk(256);
  const dim3 grd(NN / 64, NN / 64);

  // G = V^H V + I  (K = P = 128)
  zgemm_wmma<true, false><<<grd, blk, 0, stream>>>(Vr, Vi, NN, Vr, Vi, NN,
                                                   nullptr, nullptr, Rr, Ri, NN, PP);
  add_eye_k<<<2, 256, 0, stream>>>(Rr, NN);
  chol_k<<<1, 256, 0, stream>>>(Rr, Ri, NN);
  trinv_k<<<2, 256, 0, stream>>>(Rr, Ri, Ar, Ai, NN);

  // U = V^H Y  (into T buffers), RHS = A^H U
  zgemm_wmma<true, false><<<grd, blk, 0, stream>>>(Vr, Vi, NN, Yr, Yi, NN,
                                                   nullptr, nullptr, Tr, Ti, NN, PP);
  zgemm_wmma<true, false><<<grd, blk, 0, stream>>>(Ar, Ai, NN, Tr, Ti, NN,
                                                   nullptr, nullptr, Hr, Hi, NN, NN);

  // Lam_1 = RHS
  copy2_k<<<(int)(S / 256), 256, 0, stream>>>(Hr, Hi, Lr, Li, (int)S);

  // 49 remaining Stein iterations: T = A*Lam ; Lam = T*W + RHS
  for (int it = 1; it < 50; ++it) {
    zgemm_wmma<false, false><<<grd, blk, 0, stream>>>(Ar, Ai, NN, Lr, Li, NN,
                                                      nullptr, nullptr, Tr, Ti, NN, NN);
    zgemm_wmma<false, true><<<grd, blk, 0, stream>>>(Tr, Ti, NN, Wr, Wi, NN,
                                                     Hr, Hi, Lr, Li, NN, NN);
  }

  // Output: interleaved (re, im) pairs, row-major 512x512.
  interleave_k<<<(int)(S / 256), 256, 0, stream>>>(Lr, Li, (float*)d_out, (int)S);
}